// NavierStokesSimulator_7696581394717
// MI455X (gfx1250) — compile-verified
//
#include <hip/hip_runtime.h>
#include <stdint.h>

// ---------------------------------------------------------------------------
// 2048x2048 Navier-Stokes step for MI455X (gfx1250).
// Memory-bound: ~100MB working set -> fully L2-resident (192MB L2).
// FP32 VALU everywhere (no matrix math in this workload -> no WMMA).
// CDNA5 path: TDM tensor_load_to_lds DMAs (pressure 36x36 halo tile +
// divergence 34x34 tile) + s_wait_tensorcnt; 2 Jacobi iterations fused per
// launch to cut L2 traffic of the dominant phase by ~40%.
// Workspace: ~101 MB of float scratch in d_ws (+ small slack tail).
// ---------------------------------------------------------------------------

#define H  2048
#define W  2048
#define PH (H + 2)
#define PW (W + 2)
#define DT   0.01f
#define VISC 0.001f
#define JACOBI_ITERS 20

#define TILE 32
#define TIN  (TILE + 2)   // 34: intermediate / divergence tile
#define TIN2 (TILE + 4)   // 36: pressure input tile for 2 fused iterations

typedef unsigned int v4u __attribute__((ext_vector_type(4)));
typedef int          v8i __attribute__((ext_vector_type(8)));
typedef int          v4i __attribute__((ext_vector_type(4)));

#if defined(__AMDGCN__) && __has_builtin(__builtin_amdgcn_tensor_load_to_lds) && \
    __has_builtin(__builtin_amdgcn_s_wait_tensorcnt)
#define USE_TDM 1
#else
#define USE_TDM 0
#endif

__device__ __forceinline__ int iclamp(int v, int lo, int hi) {
    return v < lo ? lo : (v > hi ? hi : v);
}

// ------------------------- diffusion (explicit, edge-pad) -------------------
__global__ void diffuse_u_kernel(const float* __restrict__ u, float* __restrict__ out) {
    int x = blockIdx.x * blockDim.x + threadIdx.x;
    int y = blockIdx.y * blockDim.y + threadIdx.y;
    const int h = H + 1, w = W;
    if (x >= w || y >= h) return;
    float c = u[(size_t)y * w + x];
    float n = u[(size_t)iclamp(y - 1, 0, h - 1) * w + x];
    float s = u[(size_t)iclamp(y + 1, 0, h - 1) * w + x];
    float e = u[(size_t)y * w + iclamp(x + 1, 0, w - 1)];
    float ww = u[(size_t)y * w + iclamp(x - 1, 0, w - 1)];
    out[(size_t)y * w + x] = c + (DT * VISC) * (n + s + e + ww - 4.0f * c);
}

__device__ __forceinline__ float vb_at(const float* __restrict__ v,
                                       const float* __restrict__ dens, int y, int x) {
    // v plus buoyancy (applied before diffusion in the reference)
    float val = v[(size_t)y * (W + 1) + x];
    if (x < W) val += (DT * 0.1f) * dens[(size_t)y * W + x];
    return val;
}

__global__ void diffuse_v_kernel(const float* __restrict__ v, const float* __restrict__ dens,
                                 float* __restrict__ out) {
    int x = blockIdx.x * blockDim.x + threadIdx.x;
    int y = blockIdx.y * blockDim.y + threadIdx.y;
    const int h = H, w = W + 1;
    if (x >= w || y >= h) return;
    float c = vb_at(v, dens, y, x);
    float n = vb_at(v, dens, iclamp(y - 1, 0, h - 1), x);
    float s = vb_at(v, dens, iclamp(y + 1, 0, h - 1), x);
    float e = vb_at(v, dens, y, iclamp(x + 1, 0, w - 1));
    float ww = vb_at(v, dens, y, iclamp(x - 1, 0, w - 1));
    out[(size_t)y * w + x] = c + (DT * VISC) * (n + s + e + ww - 4.0f * c);
}

__global__ void diffuse_d_kernel(const float* __restrict__ d, float* __restrict__ out) {
    int x = blockIdx.x * blockDim.x + threadIdx.x;
    int y = blockIdx.y * blockDim.y + threadIdx.y;
    const int h = H, w = W;
    if (x >= w || y >= h) return;
    float c = d[(size_t)y * w + x];
    float n = d[(size_t)iclamp(y - 1, 0, h - 1) * w + x];
    float s = d[(size_t)iclamp(y + 1, 0, h - 1) * w + x];
    float e = d[(size_t)y * w + iclamp(x + 1, 0, w - 1)];
    float ww = d[(size_t)y * w + iclamp(x - 1, 0, w - 1)];
    out[(size_t)y * w + x] = c + (DT * VISC * 0.1f) * (n + s + e + ww - 4.0f * c);
}

// ------------------------- pressure setup ----------------------------------
// P0/P1 are (H+2)x(W+2) zero-framed: orig cell (y,x) -> padded (y+1,x+1).
__global__ void init_p_kernel(const float* __restrict__ p_in,
                              float* __restrict__ P0, float* __restrict__ P1) {
    int x = blockIdx.x * blockDim.x + threadIdx.x;
    int y = blockIdx.y * blockDim.y + threadIdx.y;
    if (x >= PW || y >= PH) return;
    float v0 = 0.0f;
    if (x >= 1 && x <= W && y >= 1 && y <= H) v0 = p_in[(size_t)(y - 1) * W + (x - 1)];
    P0[(size_t)y * PW + x] = v0;
    P1[(size_t)y * PW + x] = 0.0f;
}

// div stored into padded B at (y+1, x+1)
__global__ void div_kernel(const float* __restrict__ ud, const float* __restrict__ vd,
                           float* __restrict__ B) {
    int x = blockIdx.x * blockDim.x + threadIdx.x;
    int y = blockIdx.y * blockDim.y + threadIdx.y;
    if (x >= W || y >= H) return;
    float dv = (ud[(size_t)(y + 1) * W + x] - ud[(size_t)y * W + x] +
                vd[(size_t)y * (W + 1) + (x + 1)] - vd[(size_t)y * (W + 1) + x]) * (1.0f / DT);
    B[(size_t)(y + 1) * PW + (x + 1)] = dv;
}

// ------------------------- TDM helper --------------------------------------
#if USE_TDM
// DMA a tile_h x tile_w f32 tile of the padded PHxPW tensor into LDS.
__device__ __forceinline__ void tdm_load_2d(const float* __restrict__ gsrc,
                                            unsigned lds_off, int tile_w, int tile_h) {
    unsigned long long ga = (unsigned long long)(size_t)gsrc;
    // D# group 0: count=1 | lds_addr | global_addr[56:0] | type=2
    v4u g0;
    g0.x = 1u;
    g0.y = lds_off;
    g0.z = (unsigned)(ga & 0xFFFFFFFFu);
    g0.w = (unsigned)((ga >> 32) & 0x01FFFFFFu) | (2u << 30);
    // D# group 1: data_size=4B, tensor PWxPH, tile tile_w x tile_h, stride=PW
    const unsigned td0 = PW, td1 = PH, st0 = PW;
    v8i g1;
    g1[0] = (int)(2u << 16);                                 // data_size code 2 (4B)
    g1[1] = (int)((td0 & 0xFFFFu) << 16);                    // tensor_dim0[15:0]
    g1[2] = (int)((td0 >> 16) | ((td1 & 0xFFFFu) << 16));    // td0[31:16] | td1[15:0]
    g1[3] = (int)((td1 >> 16) | ((unsigned)tile_w << 16));   // td1[31:16] | tile_dim0
    g1[4] = (int)tile_h;                                     // tile_dim1 (tile_dim2=0)
    g1[5] = (int)st0;                                        // tensor_dim0_stride[31:0]
    g1[6] = 0;
    g1[7] = 0;
    v4i gz4 = {0, 0, 0, 0};
    v8i gz8 = {0, 0, 0, 0, 0, 0, 0, 0};
    __builtin_amdgcn_tensor_load_to_lds(g0, g1, gz4, gz4, gz8, 0);
}
#endif

// ------------- fused 2-iteration Jacobi with TDM tile staging --------------
// One workgroup produces a 32x32 tile of iteration k+2 from iteration k:
//   P tile 36x36 -> intermediate 34x34 (in LDS) -> final 32x32.
__global__ void __launch_bounds__(256) jacobi2_kernel(const float* __restrict__ Pin,
                                                      float* __restrict__ Pout,
                                                      const float* __restrict__ B) {
    __shared__ float Pt[TIN2 * TIN2];   // input pressure, origin (ty0-2, tx0-2)
    __shared__ float Bt[TIN * TIN];     // divergence,     origin (ty0-1, tx0-1)
    __shared__ float It[TIN * TIN];     // intermediate,   origin (ty0-1, tx0-1)
    const int tx0 = 2 + blockIdx.x * TILE;   // padded coords of first updated cell
    const int ty0 = 2 + blockIdx.y * TILE;

#if USE_TDM
    if (threadIdx.x < 32) {   // wave 0 issues both DMAs (in-order per wave)
        tdm_load_2d(Pin + (size_t)(ty0 - 2) * PW + (tx0 - 2),
                    (unsigned)(size_t)(&Pt[0]), TIN2, TIN2);
        tdm_load_2d(B + (size_t)(ty0 - 1) * PW + (tx0 - 1),
                    (unsigned)(size_t)(&Bt[0]), TIN, TIN);
        __builtin_amdgcn_s_wait_tensorcnt(0);
    }
    __syncthreads();
#else
    for (int i = threadIdx.x; i < TIN2 * TIN2; i += 256) {
        int r = i / TIN2, c = i % TIN2;
        size_t gy = (size_t)(ty0 - 2 + r), gx = (size_t)(tx0 - 2 + c);
        Pt[i] = (gy < PH && gx < PW) ? Pin[gy * PW + gx] : 0.0f;
    }
    for (int i = threadIdx.x; i < TIN * TIN; i += 256) {
        int r = i / TIN, c = i % TIN;
        size_t gy = (size_t)(ty0 - 1 + r), gx = (size_t)(tx0 - 1 + c);
        Bt[i] = (gy < PH && gx < PW) ? B[gy * PW + gx] : 0.0f;
    }
    __syncthreads();
#endif

    // ---- iteration 1: intermediate over 34x34 (halo ring included) ----
    for (int i = threadIdx.x; i < TIN * TIN; i += 256) {
        int r = i / TIN, c = i % TIN;
        int iy = ty0 - 1 + r, ix = tx0 - 1 + c;   // padded coords
        float val = 0.0f;                          // non-updated cells are always 0
        if (iy >= 2 && iy <= H - 1 && ix >= 2 && ix <= W - 1) {
            float up = Pt[r * TIN2 + (c + 1)];
            float dn = Pt[(r + 2) * TIN2 + (c + 1)];
            float lf = Pt[(r + 1) * TIN2 + c];
            float rt = Pt[(r + 1) * TIN2 + (c + 2)];
            val = 0.25f * (up + dn + lf + rt - Bt[i]);
        }
        It[i] = val;
    }
    __syncthreads();

    // ---- iteration 2: final 32x32 ----
#pragma unroll
    for (int k = 0; k < (TILE * TILE) / 256; ++k) {
        int idx = (int)threadIdx.x + k * 256;
        int ly = idx / TILE, lx = idx % TILE;
        int y = ty0 + ly, x = tx0 + lx;
        if (y <= H - 1 && x <= W - 1) {   // updated region: padded 2..2047
            float up = It[ly * TIN + (lx + 1)];
            float dn = It[(ly + 2) * TIN + (lx + 1)];
            float lf = It[(ly + 1) * TIN + lx];
            float rt = It[(ly + 1) * TIN + (lx + 2)];
            float b  = Bt[(ly + 1) * TIN + (lx + 1)];
            Pout[(size_t)y * PW + x] = 0.25f * (up + dn + lf + rt - b);
        }
    }
}

// ------------------------- pressure gradient -------------------------------
__global__ void grad_u_kernel(float* __restrict__ u, const float* __restrict__ P) {
    int x = blockIdx.x * blockDim.x + threadIdx.x;
    int y = blockIdx.y * blockDim.y + threadIdx.y;
    if (x >= W || y >= H + 1) return;
    if (y < 1 || y > H - 1) return;   // u[1:-1, :]
    float py  = P[(size_t)(y + 1) * PW + (x + 1)];   // p[y][x]
    float pym = P[(size_t)y * PW + (x + 1)];         // p[y-1][x]
    u[(size_t)y * W + x] -= DT * (py - pym);
}

__global__ void grad_v_kernel(float* __restrict__ v, const float* __restrict__ P) {
    int x = blockIdx.x * blockDim.x + threadIdx.x;
    int y = blockIdx.y * blockDim.y + threadIdx.y;
    if (x >= W + 1 || y >= H) return;
    if (x < 1 || x > W - 1) return;   // v[:, 1:-1]
    float px  = P[(size_t)(y + 1) * PW + (x + 1)];   // p[y][x]
    float pxm = P[(size_t)(y + 1) * PW + x];         // p[y][x-1]
    v[(size_t)y * (W + 1) + x] -= DT * (px - pxm);
}

// ------------------------- semi-Lagrangian advection -----------------------
__device__ __forceinline__ float bilerp(const float* __restrict__ f, int h, int w,
                                        float y, float x) {
    int x0 = (int)floorf(x), y0 = (int)floorf(y);
    int x1 = x0 + 1, y1 = y0 + 1;
    x0 = iclamp(x0, 0, w - 1); x1 = iclamp(x1, 0, w - 1);
    y0 = iclamp(y0, 0, h - 1); y1 = iclamp(y1, 0, h - 1);
    float fx0 = (float)x0, fx1 = (float)x1, fy0 = (float)y0, fy1 = (float)y1;
    float wa = (fx1 - x) * (fy1 - y);
    float wb = (x - fx0) * (fy1 - y);
    float wc = (fx1 - x) * (y - fy0);
    float wd = (x - fx0) * (y - fy0);
    return wa * f[(size_t)y0 * w + x0] + wb * f[(size_t)y0 * w + x1] +
           wc * f[(size_t)y1 * w + x0] + wd * f[(size_t)y1 * w + x1];
}

__global__ void advect_kernel(const float* __restrict__ field, int fh, int fw,
                              const float* __restrict__ u, const float* __restrict__ v,
                              float* __restrict__ out, float scale) {
    int x = blockIdx.x * blockDim.x + threadIdx.x;
    int y = blockIdx.y * blockDim.y + threadIdx.y;
    if (x >= fw || y >= fh) return;
    float X = (float)x, Y = (float)y;
    float x_u = fminf(fmaxf(X + 0.5f, 0.0f), (float)(W - 1));
    float u_i = bilerp(u, H + 1, W, Y, x_u);
    float y_v = fminf(fmaxf(Y + 0.5f, 0.0f), (float)(H - 1));
    float v_i = bilerp(v, H, W + 1, y_v, X);
    float px = fminf(fmaxf(X - DT * u_i, 0.0f), (float)(fw - 1));
    float py = fminf(fmaxf(Y - DT * v_i, 0.0f), (float)(fh - 1));
    out[(size_t)y * fw + x] = scale * bilerp(field, fh, fw, py, px);
}

// ------------------------- driver ------------------------------------------
extern "C" void kernel_launch(void* const* d_in, const int* in_sizes, int n_in,
                              void* d_out, int out_size, void* d_ws, size_t ws_size,
                              hipStream_t stream) {
    (void)in_sizes; (void)n_in; (void)out_size; (void)ws_size;
    const float* u_in   = (const float*)d_in[0];   // (H+1, W)
    const float* v_in   = (const float*)d_in[1];   // (H, W+1)
    const float* den_in = (const float*)d_in[2];   // (H, W)
    const float* p_in   = (const float*)d_in[3];   // (H, W)
    float* out = (float*)d_out;

    const size_t NU = (size_t)(H + 1) * W;
    const size_t NV = (size_t)H * (W + 1);
    const size_t ND = (size_t)H * W;
    const size_t NP = (size_t)PH * PW;

    float* ws  = (float*)d_ws;
    float* u_d = ws;  ws += NU;   // diffused / projected u
    float* v_d = ws;  ws += NV;   // diffused / projected v
    float* d_d = ws;  ws += ND;   // diffused density
    float* P0  = ws;  ws += NP;   // padded pressure ping
    float* P1  = ws;  ws += NP;   // padded pressure pong
    float* Bp  = ws;  ws += NP + 4 * PW;  // padded divergence (+slack so edge-tile
                                          //  TDM overreads stay in mapped scratch);
                                          //  reused as advected-u

    dim3 blk(16, 16);
    dim3 gU((W + 15) / 16, (H + 1 + 15) / 16);
    dim3 gV((W + 1 + 15) / 16, (H + 15) / 16);
    dim3 gD((W + 15) / 16, (H + 15) / 16);
    dim3 gP((PW + 15) / 16, (PH + 15) / 16);

    diffuse_u_kernel<<<gU, blk, 0, stream>>>(u_in, u_d);
    diffuse_v_kernel<<<gV, blk, 0, stream>>>(v_in, den_in, v_d);
    diffuse_d_kernel<<<gD, blk, 0, stream>>>(den_in, d_d);
    init_p_kernel<<<gP, blk, 0, stream>>>(p_in, P0, P1);
    div_kernel<<<gD, blk, 0, stream>>>(u_d, v_d, Bp);

    dim3 gJ((H + TILE - 1) / TILE, (W + TILE - 1) / TILE);  // 64x64 tiles
    float* pin = P0; float* pout = P1;
    for (int it = 0; it < JACOBI_ITERS / 2; ++it) {         // 10 fused launches
        jacobi2_kernel<<<gJ, 256, 0, stream>>>(pin, pout, Bp);
        float* t = pin; pin = pout; pout = t;
    }
    // 10 swaps -> final pressure back in P0 (== pin)

    grad_u_kernel<<<gU, blk, 0, stream>>>(u_d, pin);
    grad_v_kernel<<<gV, blk, 0, stream>>>(v_d, pin);

    float* ua = Bp;                          // div no longer needed
    float* va = (pin == P0) ? P1 : P0;       // the free pressure buffer
    advect_kernel<<<gU, blk, 0, stream>>>(u_d, H + 1, W, u_d, v_d, ua, 1.0f);
    advect_kernel<<<gV, blk, 0, stream>>>(v_d, H, W + 1, ua, v_d, va, 1.0f);
    advect_kernel<<<gD, blk, 0, stream>>>(d_d, H, W, ua, va, out, 0.995f);
}